// ProbSparseAttention_17600775979570
// MI455X (gfx1250) — compile-verified
//
#include <hip/hip_runtime.h>
#include <hip/hip_bf16.h>
#include <math.h>

// Problem constants (from reference)
#define B_  4
#define L_  2048
#define D_  512
#define H_  8
#define DK_ 64
#define U_  40      // = min(5*ceil(log(2049)), 2048)

#define KS    16    // K-slab staged through LDS per iteration
#define LDSTR 20    // padded LDS row stride (floats): 16B-aligned, bank-conflict-free

typedef float v2f __attribute__((ext_vector_type(2)));
typedef float v8f __attribute__((ext_vector_type(8)));

// ---------------------------------------------------------------------------
// GEMM: Y = X @ W^T + bias   via V_WMMA_F32_16X16X4_F32
//   X: M x K row-major, W: N x K row-major, bias: N
//   out_mode 0: Y row-major (M x N)
//   out_mode 1: Y laid out (B,H,L,dk): row m=b*L+l, col n=h*dk+d
//
// Block = 256 threads = 8 waves in a 4(M) x 2(N) grid -> 128 x 128 block tile.
// Each wave computes 32(M) x 64(N) with 8 v8f accumulators.
// K consumed in slabs of 16 through double-buffered LDS; next slab is
// prefetched to registers while the current slab feeds the WMMA pipe.
//
// WMMA f32 16x16x4 fragments (wave32):
//   A: lanes 0-15 = M rows, lane>>4 selects K pair {0,1}/{2,3} -> ds_load_b64
//   B: lanes 0-15 = N cols, lane>>4 selects K pair             -> ds_load_b64
//   D: VGPR r -> M=r (lanes 0-15), M=r+8 (lanes 16-31); lane&15 -> N col
// ---------------------------------------------------------------------------
__global__ __launch_bounds__(256)
void gemm_xwt_wmma(const float* __restrict__ X, const float* __restrict__ W,
                   const float* __restrict__ bias, float* __restrict__ Y,
                   int M, int N, int K, int out_mode) {
  __shared__ float Xs[2][128 * LDSTR];
  __shared__ float Ws[2][128 * LDSTR];

  const int tid    = threadIdx.x;
  const int lane   = tid & 31;
  const int wave   = tid >> 5;
  const int lane16 = lane & 15;
  const int khi    = lane >> 4;        // 0 or 1
  const int wm     = wave & 3;         // 0..3  (M sub-tile)
  const int wn     = wave >> 2;        // 0..1  (N sub-tile)
  const int m0     = blockIdx.x * 128;
  const int n0     = blockIdx.y * 128;

  // global prefetch mapping: 2 threads per row, 8 floats (2x float4) each
  const int pr = tid >> 1;             // 0..127
  const int pc = (tid & 1) * 8;        // 0 or 8
  const float* xg = X + (size_t)(m0 + pr) * K + pc;
  const float* wg = W + (size_t)(n0 + pr) * K + pc;

  v8f acc[2][4];
  #pragma unroll
  for (int i = 0; i < 2; ++i)
    #pragma unroll
    for (int t = 0; t < 4; ++t)
      acc[i][t] = (v8f){};

  // prefetch slab 0 and store to buffer 0
  float4 xa = *(const float4*)(xg);
  float4 xb = *(const float4*)(xg + 4);
  float4 wa = *(const float4*)(wg);
  float4 wb = *(const float4*)(wg + 4);
  {
    float* xs = &Xs[0][pr * LDSTR + pc];
    float* ws = &Ws[0][pr * LDSTR + pc];
    *(float4*)(xs) = xa; *(float4*)(xs + 4) = xb;
    *(float4*)(ws) = wa; *(float4*)(ws + 4) = wb;
  }

  const int nslab = K / KS;
  for (int s = 0; s < nslab; ++s) {
    __syncthreads();                    // buf[cur] ready; buf[nxt] free
    const int cur = s & 1;
    const int nxt = cur ^ 1;

    // issue global prefetch of next slab (overlaps with compute below)
    if (s + 1 < nslab) {
      const float* xg2 = xg + (s + 1) * KS;
      const float* wg2 = wg + (s + 1) * KS;
      xa = *(const float4*)(xg2);
      xb = *(const float4*)(xg2 + 4);
      wa = *(const float4*)(wg2);
      wb = *(const float4*)(wg2 + 4);
    }

    // consume current slab from LDS: 4 k-steps x 8 WMMA
    const float* XA = &Xs[cur][(wm * 32 + lane16) * LDSTR];
    const float* XB = XA + 16 * LDSTR;
    const float* W0 = &Ws[cur][(wn * 64 + lane16) * LDSTR];
    #pragma unroll
    for (int j = 0; j < 4; ++j) {
      const int kk = j * 4 + 2 * khi;
      v2f a0 = *(const v2f*)(XA + kk);
      v2f a1 = *(const v2f*)(XB + kk);
      v2f b0 = *(const v2f*)(W0 + kk);
      v2f b1 = *(const v2f*)(W0 + 16 * LDSTR + kk);
      v2f b2 = *(const v2f*)(W0 + 32 * LDSTR + kk);
      v2f b3 = *(const v2f*)(W0 + 48 * LDSTR + kk);
      acc[0][0] = __builtin_amdgcn_wmma_f32_16x16x4_f32(false, a0, false, b0, (short)0, acc[0][0], false, false);
      acc[0][1] = __builtin_amdgcn_wmma_f32_16x16x4_f32(false, a0, false, b1, (short)0, acc[0][1], false, false);
      acc[0][2] = __builtin_amdgcn_wmma_f32_16x16x4_f32(false, a0, false, b2, (short)0, acc[0][2], false, false);
      acc[0][3] = __builtin_amdgcn_wmma_f32_16x16x4_f32(false, a0, false, b3, (short)0, acc[0][3], false, false);
      acc[1][0] = __builtin_amdgcn_wmma_f32_16x16x4_f32(false, a1, false, b0, (short)0, acc[1][0], false, false);
      acc[1][1] = __builtin_amdgcn_wmma_f32_16x16x4_f32(false, a1, false, b1, (short)0, acc[1][1], false, false);
      acc[1][2] = __builtin_amdgcn_wmma_f32_16x16x4_f32(false, a1, false, b2, (short)0, acc[1][2], false, false);
      acc[1][3] = __builtin_amdgcn_wmma_f32_16x16x4_f32(false, a1, false, b3, (short)0, acc[1][3], false, false);
    }

    // park next slab in the other LDS buffer (last reader was iter s-1,
    // separated by the barrier at the top of this iteration)
    if (s + 1 < nslab) {
      float* xs = &Xs[nxt][pr * LDSTR + pc];
      float* ws = &Ws[nxt][pr * LDSTR + pc];
      *(float4*)(xs) = xa; *(float4*)(xs + 4) = xb;
      *(float4*)(ws) = wa; *(float4*)(ws + 4) = wb;
    }
  }

  // epilogue: bias + store
  #pragma unroll
  for (int i = 0; i < 2; ++i) {
    #pragma unroll
    for (int t = 0; t < 4; ++t) {
      const int col  = n0 + wn * 64 + t * 16 + lane16;
      const float bv = bias[col];
      #pragma unroll
      for (int r = 0; r < 8; ++r) {
        const int row = m0 + wm * 32 + i * 16 + r + 8 * khi;
        const float v = acc[i][t][r] + bv;
        if (out_mode == 0) {
          Y[(size_t)row * N + col] = v;
        } else {
          const int b = row >> 11;        // row / L_
          const int l = row & (L_ - 1);
          const int h = col >> 6;         // col / DK_
          const int d = col & (DK_ - 1);
          Y[(((size_t)(b * H_ + h) * L_ + l) << 6) + d] = v;
        }
      }
    }
  }
}

// ---------------------------------------------------------------------------
// ProbSparse sampling scores + top-U selection per (b,h).
// One block per (b,h): M[q] = max_u Q[q].K[idx[q,u]] - (1/L)*sum_u Q[q].K[idx[q,u]]
// then 40 rounds of block-wide argmax with masking.
// ---------------------------------------------------------------------------
__global__ __launch_bounds__(256)
void prob_topk_kernel(const float* __restrict__ Q, const float* __restrict__ Kmat,
                      const int* __restrict__ idx, int* __restrict__ topk) {
  __shared__ float Ms[L_];
  __shared__ float rv[256];
  __shared__ int   ri[256];
  const int bh = blockIdx.x;
  const float* Qb = Q    + (size_t)bh * L_ * DK_;
  const float* Kb = Kmat + (size_t)bh * L_ * DK_;

  for (int q = threadIdx.x; q < L_; q += 256) {
    float qr[DK_];
    #pragma unroll
    for (int d = 0; d < DK_; d += 4) {
      float4 t = *(const float4*)(Qb + (size_t)q * DK_ + d);
      qr[d] = t.x; qr[d+1] = t.y; qr[d+2] = t.z; qr[d+3] = t.w;
    }
    float mx = -__builtin_inff();
    float sm = 0.f;
    for (int u = 0; u < U_; ++u) {
      const int k = idx[q * U_ + u];
      const float* kr = Kb + (size_t)k * DK_;
      float dot = 0.f;
      #pragma unroll
      for (int d = 0; d < DK_; d += 4) {
        float4 t = *(const float4*)(kr + d);
        dot += qr[d]*t.x + qr[d+1]*t.y + qr[d+2]*t.z + qr[d+3]*t.w;
      }
      mx = fmaxf(mx, dot);
      sm += dot;
    }
    Ms[q] = mx - sm * (1.0f / (float)L_);
  }
  __syncthreads();

  for (int it = 0; it < U_; ++it) {
    float bv = -__builtin_inff(); int bi = 0;
    for (int q = threadIdx.x; q < L_; q += 256) {
      const float v = Ms[q];
      if (v > bv) { bv = v; bi = q; }
    }
    rv[threadIdx.x] = bv; ri[threadIdx.x] = bi;
    __syncthreads();
    for (int s = 128; s > 0; s >>= 1) {
      if (threadIdx.x < s) {
        if (rv[threadIdx.x + s] > rv[threadIdx.x]) {
          rv[threadIdx.x] = rv[threadIdx.x + s];
          ri[threadIdx.x] = ri[threadIdx.x + s];
        }
      }
      __syncthreads();
    }
    if (threadIdx.x == 0) {
      topk[bh * U_ + it] = ri[0];
      Ms[ri[0]] = -__builtin_inff();
    }
    __syncthreads();
  }
}

// ---------------------------------------------------------------------------
// Context init: ctx[(b*L+l)*D + h*dk + d] = mean_l V[b,h,l,d]
// One block per (b,h).
// ---------------------------------------------------------------------------
__global__ __launch_bounds__(256)
void ctx_init_mean(const float* __restrict__ V, float* __restrict__ ctx) {
  __shared__ float part[4][DK_];
  __shared__ float meanv[DK_];
  const int bh = blockIdx.x;
  const int b = bh >> 3, h = bh & 7;
  const float* Vb = V + (size_t)bh * L_ * DK_;
  const int d = threadIdx.x & 63;
  const int c = threadIdx.x >> 6;   // 0..3
  float s = 0.f;
  for (int l = c * 512; l < (c + 1) * 512; ++l)
    s += Vb[(size_t)l * DK_ + d];
  part[c][d] = s;
  __syncthreads();
  if (threadIdx.x < DK_) {
    meanv[threadIdx.x] = (part[0][threadIdx.x] + part[1][threadIdx.x] +
                          part[2][threadIdx.x] + part[3][threadIdx.x]) * (1.0f / (float)L_);
  }
  __syncthreads();
  const float mv = meanv[d];
  for (int l = c; l < L_; l += 4)
    ctx[(size_t)(b * L_ + l) * D_ + h * DK_ + d] = mv;
}

// ---------------------------------------------------------------------------
// Sparse attention over the U selected rows per (b,h):
//   scores = Q_sel @ K^T / 8, softmax over L, @ V -> scatter into ctx.
// One block per (b,h), rows processed sequentially.
// ---------------------------------------------------------------------------
__global__ __launch_bounds__(256)
void sparse_attn(const float* __restrict__ Q, const float* __restrict__ Kmat,
                 const float* __restrict__ V, const int* __restrict__ topk,
                 float* __restrict__ ctx) {
  __shared__ float sbuf[L_];
  __shared__ float qrow[DK_];
  __shared__ float red[256];
  __shared__ float part[4][DK_];
  const int bh = blockIdx.x;
  const int b = bh >> 3, h = bh & 7;
  const float* Qb = Q    + (size_t)bh * L_ * DK_;
  const float* Kb = Kmat + (size_t)bh * L_ * DK_;
  const float* Vb = V    + (size_t)bh * L_ * DK_;
  const float scale = 0.125f;  // 1/sqrt(64)

  for (int r = 0; r < U_; ++r) {
    const int q = topk[bh * U_ + r];
    if (threadIdx.x < DK_) qrow[threadIdx.x] = Qb[(size_t)q * DK_ + threadIdx.x];
    __syncthreads();

    // pass 1: scores + max
    float lmx = -__builtin_inff();
    for (int k = threadIdx.x; k < L_; k += 256) {
      const float* kr = Kb + (size_t)k * DK_;
      float dot = 0.f;
      #pragma unroll 8
      for (int d2 = 0; d2 < DK_; ++d2) dot += qrow[d2] * kr[d2];
      dot *= scale;
      sbuf[k] = dot;
      lmx = fmaxf(lmx, dot);
    }
    red[threadIdx.x] = lmx; __syncthreads();
    for (int s = 128; s > 0; s >>= 1) {
      if (threadIdx.x < s) red[threadIdx.x] = fmaxf(red[threadIdx.x], red[threadIdx.x + s]);
      __syncthreads();
    }
    const float mx = red[0];
    __syncthreads();

    // pass 2: exp + sum
    float lsum = 0.f;
    for (int k = threadIdx.x; k < L_; k += 256) {
      const float e = __expf(sbuf[k] - mx);
      sbuf[k] = e; lsum += e;
    }
    red[threadIdx.x] = lsum; __syncthreads();
    for (int s = 128; s > 0; s >>= 1) {
      if (threadIdx.x < s) red[threadIdx.x] += red[threadIdx.x + s];
      __syncthreads();
    }
    const float inv = 1.0f / red[0];

    // pass 3: ctx_d = sum_k p[k] * V[k][d]
    const int d = threadIdx.x & 63, c = threadIdx.x >> 6;
    float acc = 0.f;
    for (int k = c * 512; k < (c + 1) * 512; ++k)
      acc += sbuf[k] * Vb[(size_t)k * DK_ + d];
    part[c][d] = acc;
    __syncthreads();
    if (threadIdx.x < DK_) {
      const float tot = (part[0][threadIdx.x] + part[1][threadIdx.x] +
                         part[2][threadIdx.x] + part[3][threadIdx.x]) * inv;
      ctx[(size_t)(b * L_ + q) * D_ + h * DK_ + threadIdx.x] = tot;
    }
    __syncthreads();
  }
}

// ---------------------------------------------------------------------------
extern "C" void kernel_launch(void* const* d_in, const int* in_sizes, int n_in,
                              void* d_out, int out_size, void* d_ws, size_t ws_size,
                              hipStream_t stream) {
  const float* queries = (const float*)d_in[0];
  const float* keys    = (const float*)d_in[1];
  const float* values  = (const float*)d_in[2];
  const float* Wq = (const float*)d_in[3];
  const float* bq = (const float*)d_in[4];
  const float* Wk = (const float*)d_in[5];
  const float* bk = (const float*)d_in[6];
  const float* Wv = (const float*)d_in[7];
  const float* bv = (const float*)d_in[8];
  const float* Wo = (const float*)d_in[9];
  const float* bo = (const float*)d_in[10];
  const int* index_sample = (const int*)d_in[11];

  const size_t nQKV = (size_t)B_ * H_ * L_ * DK_;   // 4,194,304 floats each
  float* ws   = (float*)d_ws;
  float* Qp   = ws;
  float* Kp   = ws + nQKV;
  float* Vp   = ws + 2 * nQKV;
  float* CTX  = ws + 3 * nQKV;                      // (B, L, D) row-major
  int*   TOPK = (int*)(ws + 4 * nQKV);              // B*H*U ints

  const int M = B_ * L_;   // 8192
  const int N = D_;        // 512
  const int K = D_;        // 512
  dim3 gblk(M / 128, N / 128);  // (64, 4)

  // Q/K/V projections straight into (B,H,L,dk) layout
  gemm_xwt_wmma<<<gblk, 256, 0, stream>>>(queries, Wq, bq, Qp, M, N, K, 1);
  gemm_xwt_wmma<<<gblk, 256, 0, stream>>>(keys,    Wk, bk, Kp, M, N, K, 1);
  gemm_xwt_wmma<<<gblk, 256, 0, stream>>>(values,  Wv, bv, Vp, M, N, K, 1);

  // sampling scores + top-40 per (b,h)
  prob_topk_kernel<<<B_ * H_, 256, 0, stream>>>(Qp, Kp, index_sample, TOPK);

  // context = broadcast mean(V); then overwrite selected rows with attention
  ctx_init_mean<<<B_ * H_, 256, 0, stream>>>(Vp, CTX);
  sparse_attn<<<B_ * H_, 256, 0, stream>>>(Qp, Kp, Vp, TOPK, CTX);

  // output projection
  gemm_xwt_wmma<<<gblk, 256, 0, stream>>>(CTX, Wo, bo, (float*)d_out, M, N, K, 0);
}